// Mamba2Mixer_58746562674740
// MI455X (gfx1250) — compile-verified
//
#include <hip/hip_runtime.h>
#include <hip/hip_bf16.h>

#define D_MODEL   768
#define D_STATE   128
#define HEAD_DIM  64
#define CHUNK     256
#define KCONV     4
#define INTER     1536
#define NHEADS    24
#define CONV_DIM  1792          // INTER + 2*D_STATE
#define PROJ      3352          // 2*(INTER+D_STATE)+NHEADS
#define BATCH     2
#define SEQ       2048
#define NTOK      (BATCH*SEQ)   // 4096
#define NCHUNK    (SEQ/CHUNK)   // 8
#define DT_MIN    0.001f
#define DT_MAX    0.1f
#define RMS_EPS   1e-5f

typedef __attribute__((ext_vector_type(2))) float v2f;
typedef __attribute__((ext_vector_type(8))) float v8f;

__device__ __forceinline__ v8f wmma_f32(v2f a, v2f b, v8f c) {
  // D = A(16x4,f32) * B(4x16,f32) + C(16x16,f32)  -> v_wmma_f32_16x16x4_f32
  return __builtin_amdgcn_wmma_f32_16x16x4_f32(false, a, false, b, (short)0, c,
                                               false, false);
}

__device__ __forceinline__ float siluf(float x) {
  return x / (1.0f + __expf(-x));
}

#define LOAD2(dst, ptr, ofs) { (dst).x = (ptr)[ofs]; (dst).y = (ptr)[(ofs) + 1]; }

// ---------------------------------------------------------------------------
// 1) proj[4096 x 3352] = H[4096 x 768] * W_in^T   (W_in is [3352 x 768])
// 16x64 per wave, software-pipelined: next-step fragments loaded before the
// WMMAs consuming the current step, so loads stay in flight across WMMAs.
// ---------------------------------------------------------------------------
__global__ void k_proj_gemm(const float* __restrict__ H,
                            const float* __restrict__ Win,
                            float* __restrict__ proj) {
  const int lane = threadIdx.x & 31;
  const int lo = lane & 15, hi = lane >> 4;
  const int NT_N = (PROJ + 63) / 64;                     // 53 (padded)
  int tile = blockIdx.x * (blockDim.x >> 5) + (threadIdx.x >> 5);
  if (tile >= (NTOK / 16) * NT_N) return;
  int tn = tile % NT_N, tm = tile / NT_N;
  int M0 = tm * 16, N0 = tn * 64;
  const float* arow = H + (size_t)(M0 + lo) * D_MODEL;
  int col0 = N0 + lo,      col1 = N0 + 16 + lo;
  int col2 = N0 + 32 + lo, col3 = N0 + 48 + lo;
  const float* br0 = Win + (size_t)(col0 < PROJ ? col0 : PROJ - 1) * D_MODEL;
  const float* br1 = Win + (size_t)(col1 < PROJ ? col1 : PROJ - 1) * D_MODEL;
  const float* br2 = Win + (size_t)(col2 < PROJ ? col2 : PROJ - 1) * D_MODEL;
  const float* br3 = Win + (size_t)(col3 < PROJ ? col3 : PROJ - 1) * D_MODEL;
  v8f a0 = {}, a1 = {}, a2 = {}, a3 = {};
  v2f a, b0, b1, b2, b3;
  {
    int kk = 2 * hi;
    LOAD2(a,  arow, kk); LOAD2(b0, br0, kk); LOAD2(b1, br1, kk);
    LOAD2(b2, br2, kk);  LOAD2(b3, br3, kk);
  }
#pragma unroll 2
  for (int k = 0; k < D_MODEL - 4; k += 4) {
    int kn = k + 4 + 2 * hi;
    v2f an, bn0, bn1, bn2, bn3;
    LOAD2(an,  arow, kn); LOAD2(bn0, br0, kn); LOAD2(bn1, br1, kn);
    LOAD2(bn2, br2, kn);  LOAD2(bn3, br3, kn);
    a0 = wmma_f32(a, b0, a0);
    a1 = wmma_f32(a, b1, a1);
    a2 = wmma_f32(a, b2, a2);
    a3 = wmma_f32(a, b3, a3);
    a = an; b0 = bn0; b1 = bn1; b2 = bn2; b3 = bn3;
  }
  a0 = wmma_f32(a, b0, a0);
  a1 = wmma_f32(a, b1, a1);
  a2 = wmma_f32(a, b2, a2);
  a3 = wmma_f32(a, b3, a3);
#pragma unroll
  for (int r = 0; r < 8; ++r) {
    size_t ro = (size_t)(M0 + r + 8 * hi) * PROJ;
    if (col0 < PROJ) proj[ro + col0] = a0[r];
    if (col1 < PROJ) proj[ro + col1] = a1[r];
    if (col2 < PROJ) proj[ro + col2] = a2[r];
    if (col3 < PROJ) proj[ro + col3] = a3[r];
  }
}

// ---------------------------------------------------------------------------
// 2) depthwise causal conv (K=4) + bias + SiLU; split into x / B / C buffers
// ---------------------------------------------------------------------------
__global__ void k_conv(const float* __restrict__ proj,
                       const float* __restrict__ cw,
                       const float* __restrict__ cb,
                       float* __restrict__ xbuf,
                       float* __restrict__ Bbuf,
                       float* __restrict__ Cbuf) {
  long idx = (long)blockIdx.x * blockDim.x + threadIdx.x;
  if (idx >= (long)NTOK * CONV_DIM) return;
  int ch  = (int)(idx % CONV_DIM);
  int tok = (int)(idx / CONV_DIM);
  int s   = tok % SEQ;
  float acc = cb[ch];
#pragma unroll
  for (int j = 0; j < KCONV; ++j) {
    int sj = s - (KCONV - 1) + j;
    if (sj >= 0)
      acc += cw[ch * KCONV + j] *
             proj[(size_t)(tok - (KCONV - 1) + j) * PROJ + INTER + ch];
  }
  float v = siluf(acc);
  if (ch < INTER)                 xbuf[(size_t)tok * INTER + ch] = v;
  else if (ch < INTER + D_STATE)  Bbuf[(size_t)tok * D_STATE + (ch - INTER)] = v;
  else                            Cbuf[(size_t)tok * D_STATE + (ch - INTER - D_STATE)] = v;
}

// ---------------------------------------------------------------------------
// 3) dt = clip(softplus(dt_raw + dt_bias), DT_MIN, DT_MAX)
// ---------------------------------------------------------------------------
__global__ void k_dt(const float* __restrict__ proj,
                     const float* __restrict__ dt_bias,
                     float* __restrict__ dtb) {
  int idx = blockIdx.x * blockDim.x + threadIdx.x;
  if (idx >= NTOK * NHEADS) return;
  int h = idx % NHEADS, tok = idx / NHEADS;
  float x = proj[(size_t)tok * PROJ + INTER + CONV_DIM + h] + dt_bias[h];
  float sp = (x > 20.f) ? x : log1pf(__expf(x));
  dtb[idx] = fminf(fmaxf(sp, DT_MIN), DT_MAX);
}

// ---------------------------------------------------------------------------
// 4) per-(b,h,chunk) inclusive cumsum of A*dt over the 256-token chunk (LDS)
// ---------------------------------------------------------------------------
__global__ void k_cumsum(const float* __restrict__ dtb,
                         const float* __restrict__ A_log,
                         float* __restrict__ acum) {
  __shared__ float sm[CHUNK];
  int bid = blockIdx.x;                     // (b*NHEADS + h)*NCHUNK + c
  int c = bid % NCHUNK;
  int h = (bid / NCHUNK) % NHEADS;
  int b = bid / (NCHUNK * NHEADS);
  int t = threadIdx.x;
  int tok = b * SEQ + c * CHUNK + t;
  float A = -__expf(A_log[h]);
  sm[t] = A * dtb[tok * NHEADS + h];
  __syncthreads();
  for (int off = 1; off < CHUNK; off <<= 1) {
    float add = (t >= off) ? sm[t - off] : 0.f;
    __syncthreads();
    sm[t] += add;
    __syncthreads();
  }
  acum[(size_t)bid * CHUNK + t] = sm[t];
}

// ---------------------------------------------------------------------------
// 5) CB[b,c][256x256] = C_blk * B_blk^T  (K = D_STATE, head-independent)
// 16x64 per wave, software-pipelined.
// ---------------------------------------------------------------------------
__global__ void k_cb(const float* __restrict__ Cbuf,
                     const float* __restrict__ Bbuf,
                     float* __restrict__ cbm) {
  const int lane = threadIdx.x & 31, lo = lane & 15, hi = lane >> 4;
  const int TPB = (CHUNK / 16) * (CHUNK / 64);           // 64 tiles / (b,c)
  int tile = blockIdx.x * (blockDim.x >> 5) + (threadIdx.x >> 5);
  if (tile >= BATCH * NCHUNK * TPB) return;
  int bc = tile / TPB, t = tile % TPB;
  int tm = t / (CHUNK / 64), tn = t % (CHUNK / 64);
  int tokc = bc * CHUNK;                                 // b*SEQ + c*CHUNK
  int L0 = tm * 16, S0 = tn * 64;
  const float* crow = Cbuf + (size_t)(tokc + L0 + lo) * D_STATE;
  const float* br0 = Bbuf + (size_t)(tokc + S0      + lo) * D_STATE;
  const float* br1 = Bbuf + (size_t)(tokc + S0 + 16 + lo) * D_STATE;
  const float* br2 = Bbuf + (size_t)(tokc + S0 + 32 + lo) * D_STATE;
  const float* br3 = Bbuf + (size_t)(tokc + S0 + 48 + lo) * D_STATE;
  v8f a0 = {}, a1 = {}, a2 = {}, a3 = {};
  v2f a, b0, b1, b2, b3;
  {
    int kk = 2 * hi;
    LOAD2(a,  crow, kk); LOAD2(b0, br0, kk); LOAD2(b1, br1, kk);
    LOAD2(b2, br2, kk);  LOAD2(b3, br3, kk);
  }
#pragma unroll 2
  for (int k = 0; k < D_STATE - 4; k += 4) {
    int kn = k + 4 + 2 * hi;
    v2f an, bn0, bn1, bn2, bn3;
    LOAD2(an,  crow, kn); LOAD2(bn0, br0, kn); LOAD2(bn1, br1, kn);
    LOAD2(bn2, br2, kn);  LOAD2(bn3, br3, kn);
    a0 = wmma_f32(a, b0, a0);
    a1 = wmma_f32(a, b1, a1);
    a2 = wmma_f32(a, b2, a2);
    a3 = wmma_f32(a, b3, a3);
    a = an; b0 = bn0; b1 = bn1; b2 = bn2; b3 = bn3;
  }
  a0 = wmma_f32(a, b0, a0);
  a1 = wmma_f32(a, b1, a1);
  a2 = wmma_f32(a, b2, a2);
  a3 = wmma_f32(a, b3, a3);
  float* out = cbm + (size_t)bc * CHUNK * CHUNK;
#pragma unroll
  for (int r = 0; r < 8; ++r) {
    size_t ro = (size_t)(L0 + r + 8 * hi) * CHUNK;
    out[ro + S0      + lo] = a0[r];
    out[ro + S0 + 16 + lo] = a1[r];
    out[ro + S0 + 32 + lo] = a2[r];
    out[ro + S0 + 48 + lo] = a3[r];
  }
}

// ---------------------------------------------------------------------------
// 6) Y_diag[b,c,h][256x64] = (L ∘ CB) * x_disc
// 16 rows x whole head (64) per wave: masked/decayed A fragment (2 exps per
// lane per k-step) built ONCE and reused by 4 WMMAs.
// ---------------------------------------------------------------------------
__global__ void k_ydiag(const float* __restrict__ cbm,
                        const float* __restrict__ xbuf,
                        const float* __restrict__ dtb,
                        const float* __restrict__ acum,
                        float* __restrict__ y) {
  const int lane = threadIdx.x & 31, lo = lane & 15, hi = lane >> 4;
  const int TPH = CHUNK / 16;                            // 16 row-tiles
  int tile = blockIdx.x * (blockDim.x >> 5) + (threadIdx.x >> 5);
  if (tile >= BATCH * NCHUNK * NHEADS * TPH) return;
  int tm = tile % TPH; int bch = tile / TPH;
  int h = bch % NHEADS; int bc = bch / NHEADS;
  int b = bc / NCHUNK, c = bc % NCHUNK;
  int L0 = tm * 16;
  int tokc = bc * CHUNK;
  size_t cab = ((size_t)(b * NHEADS + h) * NCHUNK + c) * CHUNK;
  int lrow = L0 + lo;
  float ac_l = acum[cab + lrow];
  const float* cbrow = cbm + (size_t)bc * CHUNK * CHUNK + (size_t)lrow * CHUNK;
  const float* xh = xbuf + h * HEAD_DIM;
  v8f a0 = {}, a1 = {}, a2 = {}, a3 = {};
#pragma unroll 2
  for (int k = 0; k < CHUNK; k += 4) {
    int kk = k + 2 * hi;
    float s0 = acum[cab + kk], s1 = acum[cab + kk + 1];
    v2f a;
    a.x = (kk     <= lrow) ? cbrow[kk]     * __expf(ac_l - s0) : 0.f;
    a.y = (kk + 1 <= lrow) ? cbrow[kk + 1] * __expf(ac_l - s1) : 0.f;
    int tk0 = tokc + kk, tk1 = tk0 + 1;
    float dt0 = dtb[tk0 * NHEADS + h], dt1 = dtb[tk1 * NHEADS + h];
    const float* x0 = xh + (size_t)tk0 * INTER + lo;
    const float* x1 = xh + (size_t)tk1 * INTER + lo;
    v2f b0; b0.x = x0[0]  * dt0; b0.y = x1[0]  * dt1;
    v2f b1; b1.x = x0[16] * dt0; b1.y = x1[16] * dt1;
    v2f b2; b2.x = x0[32] * dt0; b2.y = x1[32] * dt1;
    v2f b3; b3.x = x0[48] * dt0; b3.y = x1[48] * dt1;
    a0 = wmma_f32(a, b0, a0);
    a1 = wmma_f32(a, b1, a1);
    a2 = wmma_f32(a, b2, a2);
    a3 = wmma_f32(a, b3, a3);
  }
#pragma unroll
  for (int r = 0; r < 8; ++r) {
    int l = L0 + r + 8 * hi;
    size_t ro = (size_t)(tokc + l) * INTER + h * HEAD_DIM + lo;
    y[ro]      = a0[r];
    y[ro + 16] = a1[r];
    y[ro + 32] = a2[r];
    y[ro + 48] = a3[r];
  }
}

// ---------------------------------------------------------------------------
// 7) local states[b,c,h][64x128] = (x_disc * decay)^T * B_blk  (K = CHUNK)
// 64 (p) x 16 (n) per wave: B fragment + decay factors reused by 4 WMMAs.
// ---------------------------------------------------------------------------
__global__ void k_lstate(const float* __restrict__ xbuf,
                         const float* __restrict__ dtb,
                         const float* __restrict__ Bbuf,
                         const float* __restrict__ acum,
                         float* __restrict__ ls) {
  const int lane = threadIdx.x & 31, lo = lane & 15, hi = lane >> 4;
  const int TPH = D_STATE / 16;                          // 8 col-tiles
  int tile = blockIdx.x * (blockDim.x >> 5) + (threadIdx.x >> 5);
  if (tile >= BATCH * NCHUNK * NHEADS * TPH) return;
  int tn = tile % TPH; int bch = tile / TPH;
  int h = bch % NHEADS; int bc = bch / NHEADS;
  int bq = bc / NCHUNK, c = bc % NCHUNK;
  int N0 = tn * 16;
  int tokc = bc * CHUNK;
  size_t cab = ((size_t)(bq * NHEADS + h) * NCHUNK + c) * CHUNK;
  float aend = acum[cab + CHUNK - 1];
  const float* xh = xbuf + h * HEAD_DIM;
  v8f a0 = {}, a1 = {}, a2 = {}, a3 = {};
#pragma unroll 2
  for (int k = 0; k < CHUNK; k += 4) {
    int kk = k + 2 * hi;
    int tk0 = tokc + kk, tk1 = tk0 + 1;
    float d0 = __expf(aend - acum[cab + kk])     * dtb[tk0 * NHEADS + h];
    float d1 = __expf(aend - acum[cab + kk + 1]) * dtb[tk1 * NHEADS + h];
    const float* x0 = xh + (size_t)tk0 * INTER + lo;
    const float* x1 = xh + (size_t)tk1 * INTER + lo;
    v2f f0; f0.x = x0[0]  * d0; f0.y = x1[0]  * d1;
    v2f f1; f1.x = x0[16] * d0; f1.y = x1[16] * d1;
    v2f f2; f2.x = x0[32] * d0; f2.y = x1[32] * d1;
    v2f f3; f3.x = x0[48] * d0; f3.y = x1[48] * d1;
    v2f bb;
    bb.x = Bbuf[(size_t)tk0 * D_STATE + N0 + lo];
    bb.y = Bbuf[(size_t)tk1 * D_STATE + N0 + lo];
    a0 = wmma_f32(f0, bb, a0);
    a1 = wmma_f32(f1, bb, a1);
    a2 = wmma_f32(f2, bb, a2);
    a3 = wmma_f32(f3, bb, a3);
  }
  float* out = ls + ((size_t)bc * NHEADS + h) * (HEAD_DIM * D_STATE);
#pragma unroll
  for (int r = 0; r < 8; ++r) {
    int m = r + 8 * hi;
    out[(size_t)(m)      * D_STATE + N0 + lo] = a0[r];
    out[(size_t)(m + 16) * D_STATE + N0 + lo] = a1[r];
    out[(size_t)(m + 32) * D_STATE + N0 + lo] = a2[r];
    out[(size_t)(m + 48) * D_STATE + N0 + lo] = a3[r];
  }
}

// ---------------------------------------------------------------------------
// 8) inter-chunk recurrence: P_0=0; ps[c]=P_c; P_{c+1}=exp(Aend_c)*P_c+ls[c]
// ---------------------------------------------------------------------------
__global__ void k_scan(const float* __restrict__ ls,
                       const float* __restrict__ acum,
                       float* __restrict__ ps) {
  int bh = blockIdx.x;                       // b*NHEADS + h
  int b = bh / NHEADS, h = bh % NHEADS;
  const int E = HEAD_DIM * D_STATE;          // 8192
  const int PER = E / 256;                   // 32 elems / thread
  float run[PER];
#pragma unroll
  for (int i = 0; i < PER; ++i) run[i] = 0.f;
  for (int c = 0; c < NCHUNK; ++c) {
    size_t cab = ((size_t)bh * NCHUNK + c) * CHUNK;
    float dec = __expf(acum[cab + CHUNK - 1]);
    size_t base = ((size_t)(b * NCHUNK + c) * NHEADS + h) * E;
#pragma unroll
    for (int i = 0; i < PER; ++i) {
      int e = threadIdx.x + i * 256;
      ps[base + e] = run[i];
      run[i] = dec * run[i] + ls[base + e];
    }
  }
}

// ---------------------------------------------------------------------------
// 9) Y += exp(Acum_l) * C_blk * prev_state^T + D*x   (K = D_STATE)
// 16 rows x whole head per wave, software-pipelined raw loads; the row decay
// e_l is applied at consume time.
// ---------------------------------------------------------------------------
__global__ void k_yoff(const float* __restrict__ Cbuf,
                       const float* __restrict__ ps,
                       const float* __restrict__ acum,
                       const float* __restrict__ xbuf,
                       const float* __restrict__ Dv,
                       float* __restrict__ y) {
  const int lane = threadIdx.x & 31, lo = lane & 15, hi = lane >> 4;
  const int TPH = CHUNK / 16;                            // 16 row-tiles
  int tile = blockIdx.x * (blockDim.x >> 5) + (threadIdx.x >> 5);
  if (tile >= BATCH * NCHUNK * NHEADS * TPH) return;
  int tm = tile % TPH; int bch = tile / TPH;
  int h = bch % NHEADS; int bc = bch / NHEADS;
  int b = bc / NCHUNK, c = bc % NCHUNK;
  int L0 = tm * 16;
  int tokc = bc * CHUNK;
  size_t cab = ((size_t)(b * NHEADS + h) * NCHUNK + c) * CHUNK;
  int lrow = L0 + lo;
  float e_l = __expf(acum[cab + lrow]);
  const float* crow = Cbuf + (size_t)(tokc + lrow) * D_STATE;
  const float* stb = ps + ((size_t)bc * NHEADS + h) * (HEAD_DIM * D_STATE);
  const float* st0 = stb + (size_t)(lo)      * D_STATE;  // B[k][p]=state[p][k]
  const float* st1 = stb + (size_t)(lo + 16) * D_STATE;
  const float* st2 = stb + (size_t)(lo + 32) * D_STATE;
  const float* st3 = stb + (size_t)(lo + 48) * D_STATE;
  v8f a0 = {}, a1 = {}, a2 = {}, a3 = {};
  v2f cr, b0, b1, b2, b3;
  {
    int kk = 2 * hi;
    LOAD2(cr, crow, kk); LOAD2(b0, st0, kk); LOAD2(b1, st1, kk);
    LOAD2(b2, st2, kk);  LOAD2(b3, st3, kk);
  }
#pragma unroll 2
  for (int k = 0; k < D_STATE - 4; k += 4) {
    int kn = k + 4 + 2 * hi;
    v2f crn, bn0, bn1, bn2, bn3;
    LOAD2(crn, crow, kn); LOAD2(bn0, st0, kn); LOAD2(bn1, st1, kn);
    LOAD2(bn2, st2, kn);  LOAD2(bn3, st3, kn);
    v2f a; a.x = cr.x * e_l; a.y = cr.y * e_l;
    a0 = wmma_f32(a, b0, a0);
    a1 = wmma_f32(a, b1, a1);
    a2 = wmma_f32(a, b2, a2);
    a3 = wmma_f32(a, b3, a3);
    cr = crn; b0 = bn0; b1 = bn1; b2 = bn2; b3 = bn3;
  }
  {
    v2f a; a.x = cr.x * e_l; a.y = cr.y * e_l;
    a0 = wmma_f32(a, b0, a0);
    a1 = wmma_f32(a, b1, a1);
    a2 = wmma_f32(a, b2, a2);
    a3 = wmma_f32(a, b3, a3);
  }
  float Dh = Dv[h];
#pragma unroll
  for (int r = 0; r < 8; ++r) {
    int l = L0 + r + 8 * hi;
    size_t ro = (size_t)(tokc + l) * INTER + h * HEAD_DIM + lo;
    y[ro]      = y[ro]      + a0[r] + Dh * xbuf[ro];
    y[ro + 16] = y[ro + 16] + a1[r] + Dh * xbuf[ro + 16];
    y[ro + 32] = y[ro + 32] + a2[r] + Dh * xbuf[ro + 32];
    y[ro + 48] = y[ro + 48] + a3[r] + Dh * xbuf[ro + 48];
  }
}

// ---------------------------------------------------------------------------
// 10) yf = y*silu(gate); RMS norm over INTER; yn -> (reused xbuf)
// ---------------------------------------------------------------------------
__global__ void k_norm(const float* __restrict__ y,
                       const float* __restrict__ proj,
                       const float* __restrict__ nw,
                       float* __restrict__ yn) {
  __shared__ float sv[INTER];
  __shared__ float red[256];
  int tok = blockIdx.x;
  float ss = 0.f;
  for (int i = threadIdx.x; i < INTER; i += blockDim.x) {
    float g = proj[(size_t)tok * PROJ + i];
    float v = y[(size_t)tok * INTER + i] * siluf(g);
    sv[i] = v;
    ss += v * v;
  }
  red[threadIdx.x] = ss;
  __syncthreads();
  for (int off = 128; off > 0; off >>= 1) {
    if (threadIdx.x < off) red[threadIdx.x] += red[threadIdx.x + off];
    __syncthreads();
  }
  float scale = rsqrtf(red[0] / (float)INTER + RMS_EPS);
  for (int i = threadIdx.x; i < INTER; i += blockDim.x)
    yn[(size_t)tok * INTER + i] = sv[i] * scale * nw[i];
}

// ---------------------------------------------------------------------------
// 11) out[4096 x 768] = yn[4096 x 1536] * W_out^T   (W_out is [768 x 1536])
// 16x64 per wave, software-pipelined.
// ---------------------------------------------------------------------------
__global__ void k_out_gemm(const float* __restrict__ yn,
                           const float* __restrict__ Wout,
                           float* __restrict__ out) {
  const int lane = threadIdx.x & 31, lo = lane & 15, hi = lane >> 4;
  const int NT_N = D_MODEL / 64;                          // 12
  int tile = blockIdx.x * (blockDim.x >> 5) + (threadIdx.x >> 5);
  if (tile >= (NTOK / 16) * NT_N) return;
  int tn = tile % NT_N, tm = tile / NT_N;
  int M0 = tm * 16, N0 = tn * 64;
  const float* arow = yn + (size_t)(M0 + lo) * INTER;
  const float* br0 = Wout + (size_t)(N0      + lo) * INTER;
  const float* br1 = Wout + (size_t)(N0 + 16 + lo) * INTER;
  const float* br2 = Wout + (size_t)(N0 + 32 + lo) * INTER;
  const float* br3 = Wout + (size_t)(N0 + 48 + lo) * INTER;
  v8f a0 = {}, a1 = {}, a2 = {}, a3 = {};
  v2f a, b0, b1, b2, b3;
  {
    int kk = 2 * hi;
    LOAD2(a,  arow, kk); LOAD2(b0, br0, kk); LOAD2(b1, br1, kk);
    LOAD2(b2, br2, kk);  LOAD2(b3, br3, kk);
  }
#pragma unroll 2
  for (int k = 0; k < INTER - 4; k += 4) {
    int kn = k + 4 + 2 * hi;
    v2f an, bn0, bn1, bn2, bn3;
    LOAD2(an,  arow, kn); LOAD2(bn0, br0, kn); LOAD2(bn1, br1, kn);
    LOAD2(bn2, br2, kn);  LOAD2(bn3, br3, kn);
    a0 = wmma_f32(a, b0, a0);
    a1 = wmma_f32(a, b1, a1);
    a2 = wmma_f32(a, b2, a2);
    a3 = wmma_f32(a, b3, a3);
    a = an; b0 = bn0; b1 = bn1; b2 = bn2; b3 = bn3;
  }
  a0 = wmma_f32(a, b0, a0);
  a1 = wmma_f32(a, b1, a1);
  a2 = wmma_f32(a, b2, a2);
  a3 = wmma_f32(a, b3, a3);
#pragma unroll
  for (int r = 0; r < 8; ++r) {
    size_t ro = (size_t)(M0 + r + 8 * hi) * D_MODEL;
    out[ro + N0      + lo] = a0[r];
    out[ro + N0 + 16 + lo] = a1[r];
    out[ro + N0 + 32 + lo] = a2[r];
    out[ro + N0 + 48 + lo] = a3[r];
  }
}

// ---------------------------------------------------------------------------
extern "C" void kernel_launch(void* const* d_in, const int* in_sizes, int n_in,
                              void* d_out, int out_size, void* d_ws,
                              size_t ws_size, hipStream_t stream) {
  (void)in_sizes; (void)n_in; (void)out_size; (void)ws_size;
  const float* H       = (const float*)d_in[0];   // hidden_states
  const float* Win     = (const float*)d_in[1];   // W_in
  const float* cw      = (const float*)d_in[2];   // conv_w
  const float* cbias   = (const float*)d_in[3];   // conv_b
  const float* dt_bias = (const float*)d_in[4];
  const float* A_log   = (const float*)d_in[5];
  const float* Dv      = (const float*)d_in[6];
  const float* nw      = (const float*)d_in[7];   // norm_w
  const float* Wout    = (const float*)d_in[8];
  float* out = (float*)d_out;

  float* ws    = (float*)d_ws;
  float* proj  = ws;                                                  // 13,729,792
  float* xbuf  = proj  + (size_t)NTOK * PROJ;                         //  6,291,456
  float* Bbuf  = xbuf  + (size_t)NTOK * INTER;                        //    524,288
  float* Cbuf  = Bbuf  + (size_t)NTOK * D_STATE;                      //    524,288
  float* dtbuf = Cbuf  + (size_t)NTOK * D_STATE;                      //     98,304
  float* acum  = dtbuf + (size_t)NTOK * NHEADS;                       //     98,304
  float* cbm   = acum  + (size_t)BATCH * NHEADS * NCHUNK * CHUNK;     //  1,048,576
  float* ls    = cbm   + (size_t)BATCH * NCHUNK * CHUNK * CHUNK;      //  3,145,728
  float* psb   = ls    + (size_t)BATCH * NCHUNK * NHEADS * HEAD_DIM * D_STATE;
  float* ybuf  = psb   + (size_t)BATCH * NCHUNK * NHEADS * HEAD_DIM * D_STATE;

  dim3 blk(256);   // 8 waves/block (wave32)

  { int tiles = (NTOK / 16) * ((PROJ + 63) / 64);           // 256 * 53
    k_proj_gemm<<<(tiles + 7) / 8, blk, 0, stream>>>(H, Win, proj); }
  { long n = (long)NTOK * CONV_DIM;
    k_conv<<<(unsigned)((n + 255) / 256), blk, 0, stream>>>(proj, cw, cbias,
                                                            xbuf, Bbuf, Cbuf); }
  { int n = NTOK * NHEADS;
    k_dt<<<(n + 255) / 256, blk, 0, stream>>>(proj, dt_bias, dtbuf); }
  k_cumsum<<<BATCH * NHEADS * NCHUNK, CHUNK, 0, stream>>>(dtbuf, A_log, acum);
  { int tiles = BATCH * NCHUNK * (CHUNK / 16) * (CHUNK / 64); // 16 * 64
    k_cb<<<(tiles + 7) / 8, blk, 0, stream>>>(Cbuf, Bbuf, cbm); }
  { int tiles = BATCH * NCHUNK * NHEADS * (D_STATE / 16);     // 384 * 8
    k_lstate<<<(tiles + 7) / 8, blk, 0, stream>>>(xbuf, dtbuf, Bbuf, acum, ls); }
  k_scan<<<BATCH * NHEADS, 256, 0, stream>>>(ls, acum, psb);
  { int tiles = BATCH * NCHUNK * NHEADS * (CHUNK / 16);       // 384 * 16
    k_ydiag<<<(tiles + 7) / 8, blk, 0, stream>>>(cbm, xbuf, dtbuf, acum, ybuf); }
  { int tiles = BATCH * NCHUNK * NHEADS * (CHUNK / 16);       // 384 * 16
    k_yoff<<<(tiles + 7) / 8, blk, 0, stream>>>(Cbuf, psb, acum, xbuf, Dv, ybuf); }
  k_norm<<<NTOK, 256, 0, stream>>>(ybuf, proj, nw, xbuf);     // yn reuses xbuf
  { int tiles = (NTOK / 16) * (D_MODEL / 64);                 // 256 * 12
    k_out_gemm<<<(tiles + 7) / 8, blk, 0, stream>>>(xbuf, Wout, out); }
}